// ContentAttention_74448963108993
// MI455X (gfx1250) — compile-verified
//
#include <hip/hip_runtime.h>
#include <hip/hip_bf16.h>
#include <math.h>

typedef __attribute__((ext_vector_type(2))) float v2f;
typedef __attribute__((ext_vector_type(8))) float v8f;

#define B_      128
#define R_      1024
#define RNNH_   1024
#define ATTH_   512
#define FEAT_   1024

// ---------------------------------------------------------------------------
// Kernel 1: h_att = h @ W_h2att + b  -- fp32 WMMA 16x16x4, one wave per tile.
// A (16x4 f32): lanes 0-15 hold M=0..15; VGPR0 = K=(0|2 by lane half), VGPR1 = K+1
// B (4x16 f32): lane l16 = column N; VGPR0 = row K=(0|2 by half), VGPR1 = K+1
// C (16x16 f32): VGPR i -> row M = i + 8*half, col N = l16
// ---------------------------------------------------------------------------
__global__ __launch_bounds__(32) void h2att_wmma_kernel(
    const float* __restrict__ h, const float* __restrict__ W,
    const float* __restrict__ bias, float* __restrict__ h_att)
{
    const int tm   = blockIdx.x;           // 0..7   : 16-row tile of B_
    const int tn   = blockIdx.y;           // 0..31  : 16-col tile of ATTH_
    const int lane = threadIdx.x & 31;
    const int half = lane >> 4;            // 0 or 1
    const int l16  = lane & 15;

    const float* __restrict__ arow = h + (size_t)(tm * 16 + l16) * RNNH_;
    const float* __restrict__ bcol = W + (tn * 16 + l16);

    v8f c = {0.f, 0.f, 0.f, 0.f, 0.f, 0.f, 0.f, 0.f};

    #pragma unroll 4
    for (int k = 0; k < RNNH_; k += 4) {
        const int kk = k + 2 * half;
        v2f a, b;
        a.x = arow[kk];
        a.y = arow[kk + 1];
        b.x = bcol[(size_t)kk * ATTH_];
        b.y = bcol[(size_t)(kk + 1) * ATTH_];
        c = __builtin_amdgcn_wmma_f32_16x16x4_f32(
                /*neg_a=*/false, a, /*neg_b=*/false, b,
                /*c_mod=*/(short)0, c, /*reuse_a=*/false, /*reuse_b=*/false);
    }

    const int   col = tn * 16 + l16;
    const float bv  = bias[col];
    #pragma unroll
    for (int i = 0; i < 8; ++i) {
        const int row = tm * 16 + i + 8 * half;
        h_att[(size_t)row * ATTH_ + col] = c[i] + bv;
    }
}

// ---------------------------------------------------------------------------
// Kernel 2: scores[b,r] = sum_d tanh(p_att[b,r,d] + h_att[b,d]) * w_alpha[d] + b_alpha
// One wave (32 lanes) per region row; 8 rows per 256-thread block.
// ---------------------------------------------------------------------------
__global__ __launch_bounds__(256) void scores_kernel(
    const float* __restrict__ p_att, const float* __restrict__ h_att,
    const float* __restrict__ w_alpha, const float* __restrict__ b_alpha,
    float* __restrict__ scores)
{
    __shared__ float sh[ATTH_];
    __shared__ float sw[ATTH_];

    const int b  = blockIdx.x >> 7;          // 128 blocks per batch row
    const int r0 = (blockIdx.x & 127) << 3;  // 8 regions per block
    const int t  = threadIdx.x;

    sh[t]       = h_att[(size_t)b * ATTH_ + t];
    sh[t + 256] = h_att[(size_t)b * ATTH_ + t + 256];
    sw[t]       = w_alpha[t];
    sw[t + 256] = w_alpha[t + 256];
    __syncthreads();

    const int wave = t >> 5;
    const int lane = t & 31;
    const int r    = r0 + wave;
    const float* __restrict__ prow = p_att + ((size_t)b * R_ + r) * ATTH_;

    float acc = 0.f;
    #pragma unroll
    for (int i = 0; i < 4; ++i) {
        const int d = i * 128 + lane * 4;
        const float4 p = *(const float4*)(prow + d);
        acc += tanhf(p.x + sh[d + 0]) * sw[d + 0];
        acc += tanhf(p.y + sh[d + 1]) * sw[d + 1];
        acc += tanhf(p.z + sh[d + 2]) * sw[d + 2];
        acc += tanhf(p.w + sh[d + 3]) * sw[d + 3];
    }
    #pragma unroll
    for (int off = 16; off > 0; off >>= 1)
        acc += __shfl_xor(acc, off, 32);

    if (lane == 0)
        scores[(size_t)b * R_ + r] = acc + b_alpha[0];
}

// ---------------------------------------------------------------------------
// Kernel 3: softmax over R=1024 per batch row. One 256-thread block per b.
// ---------------------------------------------------------------------------
__global__ __launch_bounds__(256) void softmax_kernel(
    const float* __restrict__ scores, float* __restrict__ weight)
{
    __shared__ float red[8];
    const int b = blockIdx.x;
    const int t = threadIdx.x;
    const int lane = t & 31;

    const float4 v = *(const float4*)(scores + (size_t)b * R_ + t * 4);

    float m = fmaxf(fmaxf(v.x, v.y), fmaxf(v.z, v.w));
    #pragma unroll
    for (int off = 16; off > 0; off >>= 1)
        m = fmaxf(m, __shfl_xor(m, off, 32));
    if (lane == 0) red[t >> 5] = m;
    __syncthreads();
    m = red[0];
    #pragma unroll
    for (int i = 1; i < 8; ++i) m = fmaxf(m, red[i]);
    __syncthreads();

    float4 e;
    e.x = expf(v.x - m); e.y = expf(v.y - m);
    e.z = expf(v.z - m); e.w = expf(v.w - m);

    float s = e.x + e.y + e.z + e.w;
    #pragma unroll
    for (int off = 16; off > 0; off >>= 1)
        s += __shfl_xor(s, off, 32);
    if (lane == 0) red[t >> 5] = s;
    __syncthreads();
    s = red[0];
    #pragma unroll
    for (int i = 1; i < 8; ++i) s += red[i];

    const float inv = 1.f / s;
    float4 o; o.x = e.x * inv; o.y = e.y * inv; o.z = e.z * inv; o.w = e.w * inv;
    *(float4*)(weight + (size_t)b * R_ + t * 4) = o;
}

// ---------------------------------------------------------------------------
// Kernel 4: partial[b,rc,f] = sum_{r in chunk rc} weight[b,r] * att_feats[b,r,f]
// grid (b=128, fchunk=4, rchunk=4); 64 threads; each thread one float4 of F.
// ---------------------------------------------------------------------------
__global__ __launch_bounds__(64) void attres_partial_kernel(
    const float* __restrict__ att_feats, const float* __restrict__ weight,
    float* __restrict__ partial)
{
    __shared__ float wsh[256];
    const int b  = blockIdx.x;
    const int fc = blockIdx.y;
    const int rc = blockIdx.z;
    const int t  = threadIdx.x;

    #pragma unroll
    for (int i = 0; i < 4; ++i)
        wsh[t + i * 64] = weight[(size_t)b * R_ + rc * 256 + t + i * 64];
    __syncthreads();

    const int f = fc * 256 + t * 4;
    const float* __restrict__ base =
        att_feats + ((size_t)b * R_ + (size_t)rc * 256) * FEAT_ + f;

    float4 acc = {0.f, 0.f, 0.f, 0.f};
    #pragma unroll 4
    for (int r = 0; r < 256; ++r) {
        const float  wv = wsh[r];
        const float4 a  = *(const float4*)(base + (size_t)r * FEAT_);
        acc.x += wv * a.x; acc.y += wv * a.y;
        acc.z += wv * a.z; acc.w += wv * a.w;
    }
    *(float4*)(partial + (((size_t)b * 4 + rc) * FEAT_) + f) = acc;
}

// ---------------------------------------------------------------------------
// Kernel 5: out[b,f] = sum_{rc=0..3} partial[b,rc,f]   (fixed order, deterministic)
// ---------------------------------------------------------------------------
__global__ __launch_bounds__(256) void attres_reduce_kernel(
    const float* __restrict__ partial, float* __restrict__ out)
{
    const int b = blockIdx.x;
    const int f = threadIdx.x * 4;
    const float* __restrict__ p = partial + (size_t)b * 4 * FEAT_ + f;

    float4 s = *(const float4*)(p);
    #pragma unroll
    for (int rc = 1; rc < 4; ++rc) {
        const float4 a = *(const float4*)(p + (size_t)rc * FEAT_);
        s.x += a.x; s.y += a.y; s.z += a.z; s.w += a.w;
    }
    *(float4*)(out + (size_t)b * FEAT_ + f) = s;
}

// ---------------------------------------------------------------------------
extern "C" void kernel_launch(void* const* d_in, const int* in_sizes, int n_in,
                              void* d_out, int out_size, void* d_ws, size_t ws_size,
                              hipStream_t stream) {
    const float* h         = (const float*)d_in[0];  // [128,1024]
    const float* att_feats = (const float*)d_in[1];  // [128,1024,1024]
    const float* p_att     = (const float*)d_in[2];  // [128,1024,512]
    const float* W_h2att   = (const float*)d_in[3];  // [1024,512]
    const float* b_h2att   = (const float*)d_in[4];  // [512]
    const float* w_alpha   = (const float*)d_in[5];  // [512]
    const float* b_alpha   = (const float*)d_in[6];  // [1]
    float* out = (float*)d_out;                      // [128,1024]

    float* ws      = (float*)d_ws;
    float* h_att   = ws;                             // 128*512   = 65536 floats
    float* scores  = h_att + (size_t)B_ * ATTH_;     // 128*1024  = 131072
    float* weight  = scores + (size_t)B_ * R_;       // 128*1024  = 131072
    float* partial = weight + (size_t)B_ * R_;       // 128*4*1024= 524288

    h2att_wmma_kernel<<<dim3(B_ / 16, ATTH_ / 16), 32, 0, stream>>>(
        h, W_h2att, b_h2att, h_att);

    scores_kernel<<<dim3(B_ * (R_ / 8)), 256, 0, stream>>>(
        p_att, h_att, w_alpha, b_alpha, scores);

    softmax_kernel<<<dim3(B_), 256, 0, stream>>>(scores, weight);

    attres_partial_kernel<<<dim3(B_, FEAT_ / 256, R_ / 256), 64, 0, stream>>>(
        att_feats, weight, partial);

    attres_reduce_kernel<<<dim3(B_), 256, 0, stream>>>(partial, out);
}